// PointerGeneratorLayer_27805618274568
// MI455X (gfx1250) — compile-verified
//
#include <hip/hip_runtime.h>
#include <cstdint>

// Problem constants from the reference: B=32, S=32, A=512, VOCAB=32000.
#define BB    32
#define SS    32
#define AA    512
#define VOCAB 32000

// Per-lane byte stride between successive unrolled rounds:
// 256 threads * 16 bytes = 4096 bytes per round.
#define ROUND_STRIDE 4096

// Fully-unrolled async global->LDS burst. INST_OFFSET shifts BOTH the LDS and
// the global address (CDNA5 ISA, async addressing), so one base address per
// lane suffices for the whole 128 KB row.
template <int K>
__device__ __forceinline__ void async_load_burst(uint32_t lds, const float* g) {
    if constexpr (K > 0) {
        async_load_burst<K - 1>(lds, g);
        asm volatile("global_load_async_to_lds_b128 %0, %1, off offset:%2 th:TH_LOAD_NT"
                     :: "v"(lds), "v"(g), "i"((K - 1) * ROUND_STRIDE)
                     : "memory");
    }
}

template <int K>
__device__ __forceinline__ void async_store_burst(float* g, uint32_t lds) {
    if constexpr (K > 0) {
        async_store_burst<K - 1>(g, lds);
        asm volatile("global_store_async_from_lds_b128 %0, %1, off offset:%2 th:TH_STORE_NT"
                     :: "v"(g), "v"(lds), "i"((K - 1) * ROUND_STRIDE)
                     : "memory");
    }
}

// One workgroup per (b,s) row: the 128 KB row lives in LDS (CDNA5: 320 KB/WGP,
// so 2 workgroups per WGP). Single HBM pass:
//   Phase 1: async DMA row global->LDS   (ASYNCcnt path, NT hint)
//   Phase 2: scatter-max attention into the LDS row via ds_max_num_f32 atomics
//   Phase 3: async DMA row LDS->global   (ASYNCcnt path, NT hint)
__launch_bounds__(256)
__global__ void pg_scatter_fused_kernel(const float* __restrict__ dec,
                                        const float* __restrict__ att,
                                        const int*   __restrict__ seq,
                                        const int*   __restrict__ table,
                                        float*       __restrict__ out) {
    __shared__ float row[VOCAB];                 // 128000 bytes of LDS

    const int bs  = blockIdx.x;                  // 0..B*S-1
    const int b   = bs >> 5;                     // bs / S
    const int tid = threadIdx.x;

    const size_t row_off = (size_t)bs * (size_t)VOCAB;
    const float* gsrc = dec + row_off;
    float*       gdst = out + row_off;

    constexpr int NV   = VOCAB / 4;              // 8000 x 16-byte transfers
    constexpr int FULL = NV / 256;               // 31 full rounds
    constexpr int REM  = NV % 256;               // 64 tail lanes

    // Per-lane base addresses (round k = byte offset k*4096 on both sides).
    const uint32_t lds0 = (uint32_t)(uintptr_t)(&row[0]) + (uint32_t)tid * 16u;
    const float*   g0s  = gsrc + (size_t)tid * 4u;
    float*         g0d  = gdst + (size_t)tid * 4u;

    // ---------------- Phase 1: global row -> LDS (async, B128, NT) ---------
    async_load_burst<FULL>(lds0, g0s);
    if (tid < REM) {
        asm volatile("global_load_async_to_lds_b128 %0, %1, off offset:%2 th:TH_LOAD_NT"
                     :: "v"(lds0), "v"(g0s), "i"(FULL * ROUND_STRIDE)
                     : "memory");
    }
#if __has_builtin(__builtin_amdgcn_s_wait_asynccnt)
    __builtin_amdgcn_s_wait_asynccnt(0);
#else
    asm volatile("s_wait_asynccnt 0" ::: "memory");
#endif
    __syncthreads();

    // ------------- Phase 2: scatter-max attention into LDS row -------------
    {
        const int*   seq_b   = seq + b * AA;
        const float* att_row = att + (size_t)bs * (size_t)AA;
        for (int a = tid; a < AA; a += 256) {
            const int   col = table[seq_b[a]];   // mapped vocab index, 0..VOCAB-1
            const float v   = att_row[a];
            uint32_t laddr = (uint32_t)(uintptr_t)(&row[col]);
            asm volatile("ds_max_num_f32 %0, %1"
                         :: "v"(laddr), "v"(v)
                         : "memory");
        }
    }
    // Make the inline-asm DS atomics visible before the barrier.
    asm volatile("s_wait_dscnt 0" ::: "memory");
    __syncthreads();

    // Reference sets out[:,:,1] = 0 AFTER the scatter-max.
    if (tid == 0) row[1] = 0.0f;
    __syncthreads();

    // ---------------- Phase 3: LDS row -> global out (async, B128, NT) -----
    async_store_burst<FULL>(g0d, lds0);
    if (tid < REM) {
        asm volatile("global_store_async_from_lds_b128 %0, %1, off offset:%2 th:TH_STORE_NT"
                     :: "v"(g0d), "v"(lds0), "i"(FULL * ROUND_STRIDE)
                     : "memory");
    }
#if __has_builtin(__builtin_amdgcn_s_wait_asynccnt)
    __builtin_amdgcn_s_wait_asynccnt(0);
#else
    asm volatile("s_wait_asynccnt 0" ::: "memory");
#endif
}

extern "C" void kernel_launch(void* const* d_in, const int* in_sizes, int n_in,
                              void* d_out, int out_size, void* d_ws, size_t ws_size,
                              hipStream_t stream) {
    (void)in_sizes; (void)n_in; (void)out_size; (void)d_ws; (void)ws_size;
    // setup_inputs() order:
    //   0: decoder_outputs (B,S,VOCAB) f32
    //   1: attention_scores (B,S,A)   f32
    //   2: input_sequence  (B,A)      i32
    //   3: repeat_idx      (S,1)      i32  (unused)
    //   4: repeat_idx2     (B,1)      i32  (unused)
    //   5: convert_table   (SRC_VOCAB,) i32
    const float* dec   = (const float*)d_in[0];
    const float* att   = (const float*)d_in[1];
    const int*   seq   = (const int*)d_in[2];
    const int*   table = (const int*)d_in[5];
    float*       out   = (float*)d_out;

    dim3 grid(BB * SS);   // 1024 workgroups, one per (b,s) row
    dim3 block(256);      // 8 wave32 waves
    pg_scatter_fused_kernel<<<grid, block, 0, stream>>>(dec, att, seq, table, out);
}